// HierarchicalClassifier_21534966022178
// MI455X (gfx1250) — compile-verified
//
#include <hip/hip_runtime.h>

typedef __attribute__((ext_vector_type(2))) float v2f;
typedef __attribute__((ext_vector_type(8))) float v8f;

#define D_DIM 2048
#define P_DIM 64
#define K_DIM 5
#define B_DIM 16384

#define KT 64            // k-tile width staged in LDS
#define ROWS_PER_WG 128  // 8 waves x 16 rows
#define PITCH 68         // floats; 272B row stride: 16B-aligned, 4-bank skew -> conflict free

// ---------------------------------------------------------------------------
// Kernel 1: parent logits = x @ parent_W^T + parent_b  via V_WMMA_F32_16X16X4_F32
// ---------------------------------------------------------------------------
__global__ __launch_bounds__(256)
void parent_gemm_wmma(const float* __restrict__ x,
                      const float* __restrict__ parent_W,
                      const float* __restrict__ parent_b,
                      float* __restrict__ parent_out)
{
    __shared__ float xs[ROWS_PER_WG * PITCH];  // 128 x 64 x-tile
    __shared__ float ws[P_DIM * PITCH];        // 64 x 64 W-tile

    const int tid  = threadIdx.x;
    const int lane = tid & 31;
    const int wave = tid >> 5;
    const int lr   = lane & 15;          // row/col within 16-lane half
    const int half = lane >> 4;          // 0 or 1
    const int kb   = half << 1;          // K offset 0 or 2 (A/B 32-bit WMMA layout)
    const int rowBase = blockIdx.x * ROWS_PER_WG;
    const int waveRow = wave << 4;       // 16 rows per wave

    v8f acc0 = {}, acc1 = {}, acc2 = {}, acc3 = {};

    for (int k0 = 0; k0 < D_DIM; k0 += KT) {
        __syncthreads();
        // Stage x tile: 128 rows x 64 floats = 2048 float4 slots, 8 per thread.
        #pragma unroll
        for (int it = 0; it < 8; ++it) {
            int slot = tid + 256 * it;
            int r = slot >> 4;
            int c = (slot & 15) << 2;
            float4 v = *(const float4*)(x + (size_t)(rowBase + r) * D_DIM + k0 + c);
            *(float4*)(&xs[r * PITCH + c]) = v;
        }
        // Stage W tile: 64 rows x 64 floats = 1024 float4 slots, 4 per thread.
        #pragma unroll
        for (int it = 0; it < 4; ++it) {
            int slot = tid + 256 * it;
            int r = slot >> 4;
            int c = (slot & 15) << 2;
            float4 v = *(const float4*)(parent_W + (size_t)r * D_DIM + k0 + c);
            *(float4*)(&ws[r * PITCH + c]) = v;
        }
        // Prefetch a slice of the next x k-tile (global_prefetch_b8).
        if (k0 + KT < D_DIM) {
            int r = tid >> 4;
            int c = (tid & 15) << 2;
            __builtin_prefetch(x + (size_t)(rowBase + r) * D_DIM + (k0 + KT) + c, 0, 1);
        }
        __syncthreads();

        // 16 WMMA k-steps of 4 over the staged tile.
        #pragma unroll
        for (int kk = 0; kk < KT; kk += 4) {
            // A (16x4 f32): lane holds row lr, K-pair at kb.
            v2f a  = *(const v2f*)(&xs[(waveRow + lr) * PITCH + kk + kb]);
            // B (4x16 f32): lane holds col lr of W^T == row lr of parent_W, K-pair at kb.
            v2f b0 = *(const v2f*)(&ws[( 0 + lr) * PITCH + kk + kb]);
            v2f b1 = *(const v2f*)(&ws[(16 + lr) * PITCH + kk + kb]);
            v2f b2 = *(const v2f*)(&ws[(32 + lr) * PITCH + kk + kb]);
            v2f b3 = *(const v2f*)(&ws[(48 + lr) * PITCH + kk + kb]);
            acc0 = __builtin_amdgcn_wmma_f32_16x16x4_f32(false, a, false, b0, (short)0, acc0, false, false);
            acc1 = __builtin_amdgcn_wmma_f32_16x16x4_f32(false, a, false, b1, (short)0, acc1, false, false);
            acc2 = __builtin_amdgcn_wmma_f32_16x16x4_f32(false, a, false, b2, (short)0, acc2, false, false);
            acc3 = __builtin_amdgcn_wmma_f32_16x16x4_f32(false, a, false, b3, (short)0, acc3, false, false);
        }
    }

    // Bias + writeback. C/D layout: VGPR j -> M = j (lanes 0-15) or j+8 (lanes 16-31), N = lr.
    const float pb0 = parent_b[ 0 + lr];
    const float pb1 = parent_b[16 + lr];
    const float pb2 = parent_b[32 + lr];
    const float pb3 = parent_b[48 + lr];
    #pragma unroll
    for (int j = 0; j < 8; ++j) {
        int row = rowBase + waveRow + j + (half << 3);
        float* o = parent_out + (size_t)row * P_DIM;
        o[ 0 + lr] = acc0[j] + pb0;
        o[16 + lr] = acc1[j] + pb1;
        o[32 + lr] = acc2[j] + pb2;
        o[48 + lr] = acc3[j] + pb3;
    }
}

// ---------------------------------------------------------------------------
// Kernel 2: per-row argmax over 64 parent logits, then 5 gathered dot products.
// One wave (32 lanes) per sample row; 8 waves per block.
// ---------------------------------------------------------------------------
__global__ __launch_bounds__(256)
void child_argmax_dot(const float* __restrict__ x,
                      const float* __restrict__ child_W,
                      const float* __restrict__ child_b,
                      const float* __restrict__ parent_logits,
                      float* __restrict__ child_out)
{
    const int lane = threadIdx.x & 31;
    const int wave = threadIdx.x >> 5;
    const int row  = blockIdx.x * 8 + wave;

    // --- argmax of parent_logits[row][0..63], first-occurrence tie-break ---
    const float* pl = parent_logits + (size_t)row * P_DIM;
    float v0 = pl[lane];
    float v1 = pl[lane + 32];
    float best = v0; int bidx = lane;
    if (v1 > best) { best = v1; bidx = lane + 32; }
    #pragma unroll
    for (int off = 16; off >= 1; off >>= 1) {
        float ov = __shfl_xor(best, off, 32);
        int   oi = __shfl_xor(bidx, off, 32);
        if (ov > best || (ov == best && oi < bidx)) { best = ov; bidx = oi; }
    }
    const int cls = bidx;  // uniform across the wave

    // --- 5 dot products of length 2048 against gathered child_W[cls] ---
    const float4* xr = (const float4*)(x + (size_t)row * D_DIM);
    const float*  wb = child_W + (size_t)cls * K_DIM * D_DIM;
    float4 a0 = {0,0,0,0}, a1 = {0,0,0,0}, a2 = {0,0,0,0}, a3 = {0,0,0,0}, a4 = {0,0,0,0};
    for (int i = lane; i < (D_DIM / 4); i += 32) {
        float4 xv = xr[i];
        float4 w0 = ((const float4*)(wb + 0 * D_DIM))[i];
        float4 w1 = ((const float4*)(wb + 1 * D_DIM))[i];
        float4 w2 = ((const float4*)(wb + 2 * D_DIM))[i];
        float4 w3 = ((const float4*)(wb + 3 * D_DIM))[i];
        float4 w4 = ((const float4*)(wb + 4 * D_DIM))[i];
        a0.x += xv.x * w0.x; a0.y += xv.y * w0.y; a0.z += xv.z * w0.z; a0.w += xv.w * w0.w;
        a1.x += xv.x * w1.x; a1.y += xv.y * w1.y; a1.z += xv.z * w1.z; a1.w += xv.w * w1.w;
        a2.x += xv.x * w2.x; a2.y += xv.y * w2.y; a2.z += xv.z * w2.z; a2.w += xv.w * w2.w;
        a3.x += xv.x * w3.x; a3.y += xv.y * w3.y; a3.z += xv.z * w3.z; a3.w += xv.w * w3.w;
        a4.x += xv.x * w4.x; a4.y += xv.y * w4.y; a4.z += xv.z * w4.z; a4.w += xv.w * w4.w;
    }
    float s0 = a0.x + a0.y + a0.z + a0.w;
    float s1 = a1.x + a1.y + a1.z + a1.w;
    float s2 = a2.x + a2.y + a2.z + a2.w;
    float s3 = a3.x + a3.y + a3.z + a3.w;
    float s4 = a4.x + a4.y + a4.z + a4.w;
    #pragma unroll
    for (int off = 16; off >= 1; off >>= 1) {
        s0 += __shfl_xor(s0, off, 32);
        s1 += __shfl_xor(s1, off, 32);
        s2 += __shfl_xor(s2, off, 32);
        s3 += __shfl_xor(s3, off, 32);
        s4 += __shfl_xor(s4, off, 32);
    }
    if (lane == 0) {
        const float* cb = child_b + (size_t)cls * K_DIM;
        float* o = child_out + (size_t)row * K_DIM;
        o[0] = s0 + cb[0];
        o[1] = s1 + cb[1];
        o[2] = s2 + cb[2];
        o[3] = s3 + cb[3];
        o[4] = s4 + cb[4];
    }
}

extern "C" void kernel_launch(void* const* d_in, const int* in_sizes, int n_in,
                              void* d_out, int out_size, void* d_ws, size_t ws_size,
                              hipStream_t stream) {
    const float* x        = (const float*)d_in[0];
    const float* parent_W = (const float*)d_in[1];
    const float* parent_b = (const float*)d_in[2];
    const float* child_W  = (const float*)d_in[3];
    const float* child_b  = (const float*)d_in[4];

    float* parent_out = (float*)d_out;                       // [B, 64]
    float* child_out  = parent_out + (size_t)B_DIM * P_DIM;  // [B, 5]

    parent_gemm_wmma<<<B_DIM / ROWS_PER_WG, 256, 0, stream>>>(x, parent_W, parent_b, parent_out);
    child_argmax_dot<<<B_DIM / 8, 256, 0, stream>>>(x, child_W, child_b, parent_out, child_out);
}